// Gaussians2D_25228637896906
// MI455X (gfx1250) — compile-verified
//
#include <hip/hip_runtime.h>
#include <math.h>

typedef __attribute__((ext_vector_type(2))) float v2f;
typedef __attribute__((ext_vector_type(8))) float v8f;

#define HDIM 1024
#define WDIM 1024
#define NG_PAD 512              // 500 gaussians padded to 32 tiles of 16
#define GTILES (NG_PAD / 16)
#define WAVES_PER_BLOCK 8
#define TPB (WAVES_PER_BLOCK * 32)

// One wave owns a 16-pixel tile (one image row, 16 consecutive columns).
// Exponent tile E[16 pix][16 gauss] = A(16x4) * B(4x16) via V_WMMA_F32_16X16X4_F32,
// then per-lane exp2 + color-weighted accumulation; cross-lane reduce at the end.
__global__ __launch_bounds__(TPB) void gaussians2d_wmma_kernel(
    const float* __restrict__ mean,   // [N,2]
    const float* __restrict__ depth,  // [N]
    const float* __restrict__ radii,  // [N]
    const float* __restrict__ color,  // [N,3]
    float* __restrict__ out,          // [3,H,W]
    int nG)
{
  __shared__ float sQ[4 * NG_PAD];  // B-matrix rows (log2-space gaussian features)
  __shared__ float sC[3 * NG_PAD];  // padded colors (0 for pad columns)

  const float LOG2E    = 1.4426950408889634f;
  const float LOG2_2PI = 2.6514961294723187f;  // log2(2*pi)

  const int tid = threadIdx.x;

  // ---- Per-block prep: gaussian features into LDS (cheap vs. 32-iter main loop) ----
  for (int g = tid; g < NG_PAD; g += TPB) {
    float q0 = 0.f, q1 = 0.f, q2 = 0.f, q3 = -128.0f;  // exp2(-128) == 0
    float c0 = 0.f, c1 = 0.f, c2 = 0.f;
    if (g < nG) {
      float m0 = mean[2 * g + 0];
      float m1 = mean[2 * g + 1];
      float d  = depth[g];
      float r  = radii[g];
      float a  = m1 - m0;
      float b  = -(m0 + m1);
      float inv_var = __expf(-2.0f * r);
      q0 = -inv_var * LOG2E;
      q1 = -inv_var * a * LOG2E;
      q2 = -inv_var * b * LOG2E;
      // log2(amp) = log2(d) - r*log2e - log2(2pi)
      q3 = -0.5f * inv_var * (a * a + b * b) * LOG2E
           + __log2f(d) - r * LOG2E - LOG2_2PI;
      c0 = color[3 * g + 0];
      c1 = color[3 * g + 1];
      c2 = color[3 * g + 2];
    }
    sQ[0 * NG_PAD + g] = q0;
    sQ[1 * NG_PAD + g] = q1;
    sQ[2 * NG_PAD + g] = q2;
    sQ[3 * NG_PAD + g] = q3;
    sC[0 * NG_PAD + g] = c0;
    sC[1 * NG_PAD + g] = c1;
    sC[2 * NG_PAD + g] = c2;
  }
  __syncthreads();

  const int wave = tid >> 5;
  const int lane = tid & 31;
  const int M    = lane & 15;   // row (pixel) index for A; column (gauss) index for B/D
  const int half = lane >> 4;   // 0: K=0,1 ; 1: K=2,3  (f32 A/B VGPR layout)

  const int T  = blockIdx.x * WAVES_PER_BLOCK + wave;  // pixel-tile id, 65536 total
  const int i  = T >> 6;                               // image row (xx)
  const int j0 = (T & 63) << 4;                        // column base (yy)

  const float x = (float)i;
  const float y = (float)(j0 + M);

  // A (16x4): P = [x^2+y^2, x-y, x+y, 1] ; lanes 0-15 carry K=0,1, lanes 16-31 carry K=2,3
  v2f A;
  A.x = half ? (x + y) : (x * x + y * y);
  A.y = half ? 1.0f    : (x - y);

  v8f acc0 = {}, acc1 = {}, acc2 = {};

  for (int t = 0; t < GTILES; ++t) {
    const int g = t * 16 + M;  // this lane's gaussian column
    v2f B;
    B.x = half ? sQ[2 * NG_PAD + g] : sQ[0 * NG_PAD + g];
    B.y = half ? sQ[3 * NG_PAD + g] : sQ[1 * NG_PAD + g];

    v8f czero = {};
    // E = A x B : 16x16 tile of log2-space exponents, full fp32
    v8f e = __builtin_amdgcn_wmma_f32_16x16x4_f32(
        false, A, false, B, (short)0, czero, false, false);

    const float c0 = sC[0 * NG_PAD + g];
    const float c1 = sC[1 * NG_PAD + g];
    const float c2 = sC[2 * NG_PAD + g];

#pragma unroll
    for (int j = 0; j < 8; ++j) {
      float gv = __builtin_amdgcn_exp2f(e[j]);  // v_exp_f32 (TRANS pipe)
      acc0[j] += c0 * gv;
      acc1[j] += c1 * gv;
      acc2[j] += c2 * gv;
    }
  }

  // Reduce over the 16 gaussian columns held by the 16 lanes of each half-group.
  // Masks 1,2,4,8 keep lanes within their 16-lane group in wave32.
#pragma unroll
  for (int m = 1; m < 16; m <<= 1) {
#pragma unroll
    for (int j = 0; j < 8; ++j) {
      acc0[j] += __shfl_xor(acc0[j], m, 32);
      acc1[j] += __shfl_xor(acc1[j], m, 32);
      acc2[j] += __shfl_xor(acc2[j], m, 32);
    }
  }

  // D layout: lanes 0-15 hold rows M=0..7 (VGPR j -> row j), lanes 16-31 rows 8..15.
  // After the reduction every lane in a half-group has identical sums, so let
  // lanes with M==c (c = channel) write channel c's 8 pixels for this half.
  if (M < 3) {
    const int   jbase = j0 + 8 * half;
    float*      o     = out + (size_t)M * ((size_t)HDIM * WDIM) + (size_t)i * WDIM + jbase;
#pragma unroll
    for (int j = 0; j < 8; ++j) {
      float v = (M == 0) ? acc0[j] : (M == 1) ? acc1[j] : acc2[j];
      o[j] = v;
    }
  }
}

extern "C" void kernel_launch(void* const* d_in, const int* in_sizes, int n_in,
                              void* d_out, int out_size, void* d_ws, size_t ws_size,
                              hipStream_t stream) {
  const float* mean  = (const float*)d_in[0];
  const float* depth = (const float*)d_in[1];
  const float* radii = (const float*)d_in[2];
  const float* color = (const float*)d_in[3];
  float*       out   = (float*)d_out;
  const int    nG    = in_sizes[1];  // number of gaussians (500)

  // 1024*1024 pixels / 16 per wave = 65536 tiles; 8 waves per block -> 8192 blocks
  const int total_tiles = (HDIM * WDIM) / 16;
  const int blocks      = total_tiles / WAVES_PER_BLOCK;

  gaussians2d_wmma_kernel<<<blocks, TPB, 0, stream>>>(mean, depth, radii, color, out, nG);
}